// SelfmixLayer_71966472012427
// MI455X (gfx1250) — compile-verified
//
#include <hip/hip_runtime.h>

// ---------------------------------------------------------------------------
// SelfmixLayer: y = tp(a,b)+tp(b,a)+k*x with a=s1e*x, b=s2e*x.
// Pure elementwise per (b,n,f) 9-group => memory-bound streaming kernel
// (~302 MB traffic -> ~13 us floor at 23.3 TB/s; ~0.19 GFLOP -> WMMA N/A).
// CDNA5 path: async global<->LDS b128 DMA (ASYNCcnt), double-buffered so
// row j+1's loads overlap row j's compute; s_wait_asynccnt with a nonzero
// per-wave threshold exploits in-order completion within loads / stores.
// ---------------------------------------------------------------------------

#define AS1 __attribute__((address_space(1)))
#define AS3 __attribute__((address_space(3)))

// Matches the builtin's parameter type: pointer to 16-byte int vector.
typedef int v4i __attribute__((vector_size(16)));

#if defined(__HIP_DEVICE_COMPILE__) &&                                   \
    __has_builtin(__builtin_amdgcn_global_load_async_to_lds_b128) &&     \
    __has_builtin(__builtin_amdgcn_global_store_async_from_lds_b128)
#define USE_ASYNC 1
#endif

#if defined(__HIP_DEVICE_COMPILE__)
#if __has_builtin(__builtin_amdgcn_s_wait_asynccnt)
#define WAIT_ASYNC(n) __builtin_amdgcn_s_wait_asynccnt(n)
#else
#define WAIT_ASYNC(n) asm volatile("s_wait_asynccnt %0" ::"i"(n) : "memory")
#endif
#else
#define WAIT_ASYNC(n) ((void)0)
#endif

// F fixed at 256 by the reference setup; one thread per feature channel.
__global__ __launch_bounds__(256) void selfmix_kernel(
    const float* __restrict__ mo, const float* __restrict__ s1g,
    const float* __restrict__ s2g, const float* __restrict__ kg,
    float* __restrict__ out, int rows) {
  constexpr int F          = 256;
  constexpr int GROUP      = 9;
  constexpr int ROW_FLOATS = F * GROUP;      // 2304 floats = 9216 B per (b,n) row
  constexpr int ROW_VEC4   = ROW_FLOATS / 4; // 576 b128 transfers per row
  constexpr int ROWS_PER_WG = 8;
  // 576 = 2*256 + 64: threads <64 (waves 0-1) issue 3 b128s/row, rest issue 2.

  __shared__ __align__(16) float lin[2][ROW_FLOATS];   // double-buffered in
  __shared__ __align__(16) float lout[2][ROW_FLOATS];  // double-buffered out

  const int tid = threadIdx.x;
  const int f   = tid;

  // Per-feature scalars (s1, s2, k are (F,3)); derived coefficients after
  // collapsing tp(a,b)+tp(b,a): cross terms cancel, sym_l2 is symmetric.
  const float a0 = s1g[3 * f + 0], a1 = s1g[3 * f + 1], a2 = s1g[3 * f + 2];
  const float b0 = s2g[3 * f + 0], b1 = s2g[3 * f + 1], b2 = s2g[3 * f + 2];
  const float k0 = kg[3 * f + 0], k1 = kg[3 * f + 1], k2 = kg[3 * f + 2];
  const float c00 = 2.0f * a0 * b0;              // x0^2 weight in out0
  const float c01 = 2.0f * a1 * b1;              // |x1|^2 in out0, sym weight in out2
  const float c02 = 2.0f * a2 * b2;              // |x2|^2 weight in out0
  const float c1  = 2.0f * (a0 * b1 + b0 * a1);  // x0*x1 weight in out1
  const float c2  = 2.0f * (a0 * b2 + b0 * a2);  // x0*x2 weight in out2
  const float SQRT3 = 1.7320508075688772f;

  const int row0  = blockIdx.x * ROWS_PER_WG;
  int       nrows = rows - row0;
  if (nrows <= 0) return;
  if (nrows > ROWS_PER_WG) nrows = ROWS_PER_WG;

  // ---- helpers as lambdas so async/fallback share one pipeline shape ----
  auto stage_in = [&](int buf, int j) {
    const float* g = mo + (size_t)(row0 + j) * ROW_FLOATS;
    float*       l = lin[buf];
#ifdef USE_ASYNC
    for (int i = tid; i < ROW_VEC4; i += 256)
      __builtin_amdgcn_global_load_async_to_lds_b128(
          (AS1 v4i*)(g + (size_t)i * 4), (AS3 v4i*)(l + i * 4), 0, 0);
#else
    for (int i = tid; i < ROW_VEC4; i += 256)
      ((float4*)l)[i] = ((const float4*)g)[i];
#endif
  };
  auto stage_out = [&](int buf, int j) {
    float*       g = out + (size_t)(row0 + j) * ROW_FLOATS;
    const float* l = lout[buf];
#ifdef USE_ASYNC
    for (int i = tid; i < ROW_VEC4; i += 256)
      __builtin_amdgcn_global_store_async_from_lds_b128(
          (AS1 v4i*)(g + (size_t)i * 4), (AS3 v4i*)(l + i * 4), 0, 0);
#else
    for (int i = tid; i < ROW_VEC4; i += 256)
      ((float4*)g)[i] = ((const float4*)l)[i];
#endif
  };

  stage_in(0, 0);  // prologue prefetch

  for (int j = 0; j < nrows; ++j) {
    const int cur = j & 1;

    if (j + 1 < nrows) {
      stage_in(cur ^ 1, j + 1);
#ifdef USE_ASYNC
      // Wait asynccnt <= L (L = this wave's transfers/row). In-order
      // completion within loads and within stores => loads(j) and
      // stores(j-2) (same lout buffer) are provably retired, while the
      // L just-issued prefetch loads may remain in flight.
      if (tid < 64) { WAIT_ASYNC(3); } else { WAIT_ASYNC(2); }
#endif
    } else {
#ifdef USE_ASYNC
      WAIT_ASYNC(0);  // no newer ops to hide behind: drain everything
#endif
    }
    __syncthreads();

    // ---- compute: thread t owns group f=t (stride 9 ⟂ 64 LDS banks) ----
    const float* li = lin[cur];
    float*       lo = lout[cur];
    const int    t9 = f * GROUP;
    const float x0 = li[t9 + 0];
    const float p = li[t9 + 1], q = li[t9 + 2], r = li[t9 + 3];
    const float v0 = li[t9 + 4], v1 = li[t9 + 5], v2 = li[t9 + 6];
    const float v3 = li[t9 + 7], v4 = li[t9 + 8];

    const float n1 = p * p + q * q + r * r;
    const float n2 = v0 * v0 + v1 * v1 + v2 * v2 + v3 * v3 + v4 * v4;

    const float y0 = c00 * x0 * x0 + c01 * n1 + c02 * n2 + k0 * x0;
    const float t1 = c1 * x0 + k1;
    const float t2 = c2 * x0 + k2;
    const float s0v = 2.0f * p * q;
    const float s1v = 2.0f * q * r;
    const float s2v = (2.0f * r * r - p * p - q * q) / SQRT3;
    const float s3v = 2.0f * p * r;
    const float s4v = p * p - q * q;

    lo[t9 + 0] = y0;
    lo[t9 + 1] = t1 * p;
    lo[t9 + 2] = t1 * q;
    lo[t9 + 3] = t1 * r;
    lo[t9 + 4] = t2 * v0 + c01 * s0v;
    lo[t9 + 5] = t2 * v1 + c01 * s1v;
    lo[t9 + 6] = t2 * v2 + c01 * s2v;
    lo[t9 + 7] = t2 * v3 + c01 * s3v;
    lo[t9 + 8] = t2 * v4 + c01 * s4v;
    __syncthreads();

    stage_out(cur, j);
  }
#ifdef USE_ASYNC
  WAIT_ASYNC(0);  // drain trailing stores (S_ENDPGM also implies wait-idle)
#endif
}

extern "C" void kernel_launch(void* const* d_in, const int* in_sizes, int n_in,
                              void* d_out, int out_size, void* d_ws, size_t ws_size,
                              hipStream_t stream) {
  (void)n_in; (void)out_size; (void)d_ws; (void)ws_size;
  const float* mo = (const float*)d_in[0];
  const float* s1 = (const float*)d_in[1];
  const float* s2 = (const float*)d_in[2];
  const float* k  = (const float*)d_in[3];
  float* out = (float*)d_out;

  const int F = in_sizes[1] / 3;                              // 256
  const long long rows = (long long)in_sizes[0] / (9LL * F);  // 16 * 1024 = 16384
  const int ROWS_PER_WG = 8;
  const int grid = (int)((rows + ROWS_PER_WG - 1) / ROWS_PER_WG);  // 2048 WGs

  selfmix_kernel<<<grid, 256, 0, stream>>>(mo, s1, s2, k, out, (int)rows);
}